// GCNModel_88313117541056
// MI455X (gfx1250) — compile-verified
//
#include <hip/hip_runtime.h>
#include <math.h>

typedef __attribute__((ext_vector_type(2))) float v2f;
typedef __attribute__((ext_vector_type(8))) float v8f;

#define THREADS 256

__device__ __forceinline__ void atomAddF(float* p, float v) {
    __hip_atomic_fetch_add(p, v, __ATOMIC_RELAXED, __HIP_MEMORY_SCOPE_AGENT);
}

// ---------------- degree / normalization ----------------
__global__ void k_init_deg(float* __restrict__ deg, int N) {
    int i = blockIdx.x * THREADS + threadIdx.x;
    if (i < N) deg[i] = 1.0f;   // self loop
}

__global__ void k_deg_accum(const int* __restrict__ dst, float* __restrict__ deg, int E) {
    int e = blockIdx.x * THREADS + threadIdx.x;
    if (e < E) atomAddF(&deg[dst[e]], 1.0f);
}

__global__ void k_rsqrt(float* __restrict__ deg, int N) {
    int i = blockIdx.x * THREADS + threadIdx.x;
    if (i < N) deg[i] = rsqrtf(deg[i]);
}

// ---------------- WMMA GEMM: C[MxN] = A[Mx128] * B[128xN] ----------------
// Block = 256 threads = 8 waves. Each block: 128 rows x 64 cols.
// B staged in LDS as k-pair-interleaved float2 so each lane's WMMA B fragment
// is ONE aligned ds_load_b64 (no VGPR re-pack movs). Pair-row stride 80
// float2 (160 dwords == 32 mod 64 banks) -> half-waves hit disjoint bank
// halves: conflict-free b64 reads.
#define BP_STRIDE 80
__global__ void k_gemm128(const float* __restrict__ A, const float* __restrict__ B,
                          float* __restrict__ C, int M, int N) {
    __shared__ float2 Bs[64 * BP_STRIDE];   // 40 KB
    const int colBase = blockIdx.y * 64;

    for (int t = threadIdx.x; t < 64 * 64; t += THREADS) {
        int pk = t >> 6, n = t & 63;        // pk = k/2 pair index
        float2 p;
        p.x = B[(size_t)(2 * pk    ) * N + colBase + n];
        p.y = B[(size_t)(2 * pk + 1) * N + colBase + n];
        Bs[pk * BP_STRIDE + n] = p;
    }
    __syncthreads();

    const int wave = threadIdx.x >> 5;
    const int lane = threadIdx.x & 31;
    const int rowTile = blockIdx.x * 8 + wave;
    if (rowTile * 16 >= M) return;   // M is a multiple of 16 here (100000)

    const int r  = lane & 15;   // row within A tile / col within B,C tile
    const int kh = lane >> 4;   // which K-half of the 16x4 / 4x16 fragment

    const float* __restrict__ Arow = A + (size_t)(rowTile * 16 + r) * 128;

    v8f acc[4];
    #pragma unroll
    for (int t = 0; t < 4; ++t) acc[t] = v8f{0.f,0.f,0.f,0.f,0.f,0.f,0.f,0.f};

    for (int k = 0; k < 128; k += 4) {
        const int kk = k + 2 * kh;          // even
        const int pk = (k >> 1) + kh;       // pair-row in Bs
        v2f a = *(const v2f*)(Arow + kk);   // A[row][kk], A[row][kk+1]
        #pragma unroll
        for (int t = 0; t < 4; ++t) {
            v2f b = *(const v2f*)&Bs[pk * BP_STRIDE + t * 16 + r]; // ds_load_b64
            acc[t] = __builtin_amdgcn_wmma_f32_16x16x4_f32(
                false, a, false, b, (short)0, acc[t], false, false);
        }
    }

    #pragma unroll
    for (int t = 0; t < 4; ++t) {
        #pragma unroll
        for (int v = 0; v < 8; ++v) {
            int row = rowTile * 16 + v + 8 * kh;
            C[(size_t)row * N + colBase + t * 16 + r] = acc[t][v];
        }
    }
}

// ---------------- self-loop init: Aout = T * dinv[node]^2 ----------------
__global__ void k_selfinit(const float* __restrict__ T, const float* __restrict__ dinv,
                           float* __restrict__ Aout, long total4, int dq /* D/4 */) {
    long idx = (long)blockIdx.x * THREADS + threadIdx.x;
    if (idx >= total4) return;
    int node = (int)(idx / dq);
    float s = dinv[node];
    s = s * s;
    float4 v = ((const float4*)T)[idx];
    v.x *= s; v.y *= s; v.z *= s; v.w *= s;
    ((float4*)Aout)[idx] = v;
}

// ---------------- edge scatter, D=128 (wave per edge, float4 per lane) ----------------
__global__ void k_scatter128(const float* __restrict__ T, const float* __restrict__ dinv,
                             const int* __restrict__ src, const int* __restrict__ dst,
                             float* __restrict__ A, int E) {
    const int wave = threadIdx.x >> 5;
    const int lane = threadIdx.x & 31;
    const int e = blockIdx.x * 8 + wave;
    if (e >= E) return;
    const int s = src[e], d = dst[e];
    const float w = dinv[s] * dinv[d];
    float4 v = ((const float4*)(T + (size_t)s * 128))[lane];
    float* Ad = A + (size_t)d * 128 + lane * 4;
    atomAddF(Ad + 0, v.x * w);
    atomAddF(Ad + 1, v.y * w);
    atomAddF(Ad + 2, v.z * w);
    atomAddF(Ad + 3, v.w * w);
}

// ---------------- edge scatter, D=64 (wave per edge, float2 per lane) ----------------
__global__ void k_scatter64(const float* __restrict__ T, const float* __restrict__ dinv,
                            const int* __restrict__ src, const int* __restrict__ dst,
                            float* __restrict__ A, int E) {
    const int wave = threadIdx.x >> 5;
    const int lane = threadIdx.x & 31;
    const int e = blockIdx.x * 8 + wave;
    if (e >= E) return;
    const int s = src[e], d = dst[e];
    const float w = dinv[s] * dinv[d];
    float2 v = ((const float2*)(T + (size_t)s * 64))[lane];
    float* Ad = A + (size_t)d * 64 + lane * 2;
    atomAddF(Ad + 0, v.x * w);
    atomAddF(Ad + 1, v.y * w);
}

// ---------------- bias + relu (D=128) ----------------
__global__ void k_bias_relu(float* __restrict__ A, const float* __restrict__ b, long total) {
    long idx = (long)blockIdx.x * THREADS + threadIdx.x;
    if (idx >= total) return;
    float v = A[idx] + b[idx & 127];
    A[idx] = v > 0.f ? v : 0.f;
}

// ---------------- bias + log_softmax (D=64, wave per node, in place) ----------------
__global__ void k_bias_logsoftmax(float* __restrict__ O, const float* __restrict__ b, int N) {
    const int wave = threadIdx.x >> 5;
    const int lane = threadIdx.x & 31;
    const int i = blockIdx.x * 8 + wave;
    if (i >= N) return;
    float* row = O + (size_t)i * 64;
    float x0 = row[lane]      + b[lane];
    float x1 = row[lane + 32] + b[lane + 32];
    float m = fmaxf(x0, x1);
    #pragma unroll
    for (int off = 16; off > 0; off >>= 1) m = fmaxf(m, __shfl_xor(m, off, 32));
    float s = __expf(x0 - m) + __expf(x1 - m);
    #pragma unroll
    for (int off = 16; off > 0; off >>= 1) s += __shfl_xor(s, off, 32);
    float l = __logf(s);
    row[lane]      = x0 - m - l;
    row[lane + 32] = x1 - m - l;
}

extern "C" void kernel_launch(void* const* d_in, const int* in_sizes, int n_in,
                              void* d_out, int out_size, void* d_ws, size_t ws_size,
                              hipStream_t stream) {
    const float* x  = (const float*)d_in[0];
    const int*   ei = (const int*)  d_in[1];
    const float* W1 = (const float*)d_in[2];
    const float* b1 = (const float*)d_in[3];
    const float* W2 = (const float*)d_in[4];
    const float* b2 = (const float*)d_in[5];
    float* out = (float*)d_out;

    const int N = in_sizes[0] / 128;   // 100000
    const int E = in_sizes[1] / 2;     // 1600000
    const int* src = ei;
    const int* dst = ei + E;

    // workspace layout (floats): dinv[N] | T1[N*128] | A1[N*128] | T2[N*64]
    float* dinv = (float*)d_ws;
    float* T1   = dinv + N;
    float* A1   = T1 + (size_t)N * 128;
    float* T2   = A1 + (size_t)N * 128;

    dim3 blk(THREADS);
    const int Mtiles = (N + 15) / 16;

    // normalization
    k_init_deg <<<(N + THREADS - 1) / THREADS, blk, 0, stream>>>(dinv, N);
    k_deg_accum<<<(E + THREADS - 1) / THREADS, blk, 0, stream>>>(dst, dinv, E);
    k_rsqrt    <<<(N + THREADS - 1) / THREADS, blk, 0, stream>>>(dinv, N);

    // layer 1: T1 = X @ W1
    dim3 g1((Mtiles + 7) / 8, 2);  // N=128 -> two 64-col panels
    k_gemm128<<<g1, blk, 0, stream>>>(x, W1, T1, N, 128);

    long tot1_4 = (long)N * 32;
    k_selfinit<<<(unsigned)((tot1_4 + THREADS - 1) / THREADS), blk, 0, stream>>>(T1, dinv, A1, tot1_4, 32);
    k_scatter128<<<(E + 7) / 8, blk, 0, stream>>>(T1, dinv, src, dst, A1, E);
    k_bias_relu<<<(unsigned)(((long)N * 128 + THREADS - 1) / THREADS), blk, 0, stream>>>(A1, b1, (long)N * 128);

    // layer 2: T2 = H @ W2
    dim3 g2((Mtiles + 7) / 8, 1);  // N=64 -> one panel
    k_gemm128<<<g2, blk, 0, stream>>>(A1, W2, T2, N, 64);

    long tot2_4 = (long)N * 16;
    k_selfinit<<<(unsigned)((tot2_4 + THREADS - 1) / THREADS), blk, 0, stream>>>(T2, dinv, out, tot2_4, 16);
    k_scatter64<<<(E + 7) / 8, blk, 0, stream>>>(T2, dinv, src, dst, out, E);

    k_bias_logsoftmax<<<(N + 7) / 8, blk, 0, stream>>>(out, b2, N);
}